// LinkerEncoder_42296837931533
// MI455X (gfx1250) — compile-verified
//
#include <hip/hip_runtime.h>
#include <cstddef>
#include <cstdint>

typedef __attribute__((ext_vector_type(2))) float v2f;
typedef __attribute__((ext_vector_type(8))) float v8f;

#define NNODES 50000
#define NEDGES 600000
#define NGRAPH 1000

// ---------------- utility ----------------
__global__ void fill_f32(float* __restrict__ p, int n, float v) {
  int i = blockIdx.x * blockDim.x + threadIdx.x;
  if (i < n) p[i] = v;
}

// deg[col] += 1 for every edge (deg pre-filled with 1.0 for the self loop)
__global__ void edge_deg(const int* __restrict__ ei, int E, float* __restrict__ deg) {
  int e = blockIdx.x * blockDim.x + threadIdx.x;
  if (e < E) atomicAdd(&deg[ei[E + e]], 1.0f);
}

__global__ void deg_to_dinv(float* __restrict__ d, int n) {
  int i = blockIdx.x * blockDim.x + threadIdx.x;
  if (i < n) {
    float v = d[i];
    d[i] = (v > 0.0f) ? rsqrtf(fmaxf(v, 1e-12f)) : 0.0f;
  }
}

// ---------------- WMMA f32 GEMM:  H[M,N] = X[M,K] @ W[K,N] (+bias)(+relu) ----------------
// N = NT*16. One wave owns a 16-row stripe and all NT column tiles.
// V_WMMA_F32_16X16X4_F32 layouts (wave32):
//   A 16x4 : lane<16 -> row=lane, k={0,1} in v[0],v[1]; lane>=16 -> row=lane-16, k={2,3}
//   B 4x16 : lane<16 -> col=lane, k={0,1};              lane>=16 -> col=lane-16, k={2,3}
//   C/D    : col=lane&15, row = (lane>>4)*8 + vgpr
// BCHK=false requires M % 16 == 0 (no per-lane row guards in the hot loop).
template <int NT, bool BCHK>
__global__ void gemm_wmma_f32(const float* __restrict__ X, const float* __restrict__ W,
                              const float* __restrict__ bias, float* __restrict__ H,
                              int M, int K, int relu) {
  constexpr int N = NT * 16;
  extern __shared__ float sW[];                 // Kp x N, zero-padded in K
  const int Kp = (K + 3) & ~3;
  for (int idx = threadIdx.x; idx < Kp * N; idx += blockDim.x) {
    int k = idx / N, n = idx - k * N;
    sW[idx] = (k < K) ? W[k * N + n] : 0.0f;
  }
  __syncthreads();

  const int lane = threadIdx.x & 31;
  const int wave = threadIdx.x >> 5;
  const int m0 = (blockIdx.x * 8 + wave) * 16;  // 8 waves per 256-thread block
  if (m0 >= M) return;                          // uniform per wave -> EXEC stays all-1s

  const int nlane = lane & 15;
  const int koff  = (lane >> 4) * 2;
  const int mA    = m0 + (lane & 15);
  const bool aok  = !BCHK || (mA < M);
  const float* __restrict__ Xrow = X + (size_t)mA * K;

  v8f acc[NT] = {};

  auto mma_step = [&](v2f a, int k) {
#pragma unroll
    for (int t = 0; t < NT; ++t) {
      v2f b;
      b.x = sW[(k    ) * N + t * 16 + nlane];
      b.y = sW[(k + 1) * N + t * 16 + nlane];
      acc[t] = __builtin_amdgcn_wmma_f32_16x16x4_f32(false, a, false, b,
                                                     (short)0, acc[t], false, false);
    }
  };

  const int Kmain = K & ~3;
  // main loop: k and k+1 always < K, only row guard (and only when BCHK)
  for (int k0 = 0; k0 < Kmain; k0 += 4) {
    const int k = k0 + koff;
    v2f a;
    if (BCHK) {
      a.x = aok ? Xrow[k]     : 0.0f;
      a.y = aok ? Xrow[k + 1] : 0.0f;
    } else {
      a.x = Xrow[k];
      a.y = Xrow[k + 1];
    }
    mma_step(a, k);
  }
  // K tail (only when K % 4 != 0): per-element K guard
  if (Kmain < Kp) {
    const int k = Kmain + koff;
    v2f a;
    a.x = (aok && (k     < K)) ? Xrow[k]     : 0.0f;
    a.y = (aok && (k + 1 < K)) ? Xrow[k + 1] : 0.0f;
    mma_step(a, k);
  }

  const int mrow = m0 + (lane >> 4) * 8;
#pragma unroll
  for (int t = 0; t < NT; ++t) {
    const int n = t * 16 + nlane;
    const float bv = bias ? bias[n] : 0.0f;
#pragma unroll
    for (int v = 0; v < 8; ++v) {
      const int m = mrow + v;
      float val = acc[t][v] + bv;
      if (relu) val = fmaxf(val, 0.0f);
      if (!BCHK || (m < M)) H[(size_t)m * N + n] = val;
    }
  }
}

// ---------------- GCN aggregation ----------------
// out[i][f] = dinv[i]^2 * H[i][f]   (self-loop term; also fully initializes out)
__global__ void self_init(const float* __restrict__ H, const float* __restrict__ dinv,
                          float* __restrict__ out, int n, int N) {
  int idx = blockIdx.x * blockDim.x + threadIdx.x;
  if (idx < n * N) {
    int i = idx / N;
    float dv = dinv[i];
    out[idx] = dv * dv * H[idx];
  }
}

// out[c] += dinv[r]*dinv[c] * H[r]   (4 features per thread, float4 gather; L2-resident)
__global__ void edge_scatter(const int* __restrict__ ei, const float* __restrict__ dinv,
                             const float* __restrict__ H, float* __restrict__ out,
                             int E, int N) {
  const int nch = N >> 2;
  int idx = blockIdx.x * blockDim.x + threadIdx.x;
  int e = idx / nch;
  if (e >= E) return;
  int ch = idx - e * nch;
  int r = ei[e];
  int c = ei[E + e];
  float nrm = dinv[r] * dinv[c];
  float4 hv = ((const float4*)(H + (size_t)r * N))[ch];
  float* dst = out + (size_t)c * N + ch * 4;
  atomicAdd(dst + 0, nrm * hv.x);
  atomicAdd(dst + 1, nrm * hv.y);
  atomicAdd(dst + 2, nrm * hv.z);
  atomicAdd(dst + 3, nrm * hv.w);
}

__global__ void bias_relu(float* __restrict__ h, const float* __restrict__ b, int n, int N) {
  int idx = blockIdx.x * blockDim.x + threadIdx.x;
  if (idx < n * N) h[idx] = fmaxf(h[idx] + b[idx & (N - 1)], 0.0f);
}

// ---------------- global mean pool ----------------
__global__ void pool_sum(const float* __restrict__ H, const int* __restrict__ batch,
                         float* __restrict__ pooled, int n) {
  int idx = blockIdx.x * blockDim.x + threadIdx.x;
  if (idx < n * 64) {
    int i = idx >> 6, f = idx & 63;
    atomicAdd(&pooled[batch[i] * 64 + f], H[(size_t)i * 64 + f]);
  }
}

__global__ void pool_cnt(const int* __restrict__ batch, float* __restrict__ cnt, int n) {
  int i = blockIdx.x * blockDim.x + threadIdx.x;
  if (i < n) atomicAdd(&cnt[batch[i]], 1.0f);
}

__global__ void pool_div(float* __restrict__ pooled, const float* __restrict__ cnt, int g) {
  int idx = blockIdx.x * blockDim.x + threadIdx.x;
  if (idx < g * 64) pooled[idx] /= fmaxf(cnt[idx >> 6], 1.0f);
}

// ---------------- LayerNorm over last dim (64) ----------------
__global__ void layernorm_k(const float* __restrict__ g, const float* __restrict__ gamma,
                            const float* __restrict__ beta, float* __restrict__ out, int rows) {
  int r = blockIdx.x * blockDim.x + threadIdx.x;
  if (r >= rows) return;
  const float* row = g + (size_t)r * 64;
  float mu = 0.0f;
  for (int f = 0; f < 64; ++f) mu += row[f];
  mu *= (1.0f / 64.0f);
  float var = 0.0f;
  for (int f = 0; f < 64; ++f) { float d = row[f] - mu; var += d * d; }
  var *= (1.0f / 64.0f);
  float inv = rsqrtf(var + 1e-5f);
  for (int f = 0; f < 64; ++f)
    out[(size_t)r * 64 + f] = (row[f] - mu) * inv * gamma[f] + beta[f];
}

// ---------------- driver ----------------
extern "C" void kernel_launch(void* const* d_in, const int* in_sizes, int n_in,
                              void* d_out, int out_size, void* d_ws, size_t ws_size,
                              hipStream_t stream) {
  const float* x     = (const float*)d_in[0];
  const int*   ei    = (const int*)d_in[1];
  const int*   batch = (const int*)d_in[2];
  const float* W1 = (const float*)d_in[3];
  const float* b1 = (const float*)d_in[4];
  const float* W2 = (const float*)d_in[5];
  const float* b2 = (const float*)d_in[6];
  const float* W3 = (const float*)d_in[7];
  const float* b3 = (const float*)d_in[8];
  const float* Wf1 = (const float*)d_in[9];
  const float* bf1 = (const float*)d_in[10];
  const float* Wf2 = (const float*)d_in[11];
  const float* bf2 = (const float*)d_in[12];
  const float* gamma = (const float*)d_in[13];
  const float* beta  = (const float*)d_in[14];
  float* out = (float*)d_out;

  char* ws = (char*)d_ws;
  size_t off = 0;
  auto walloc = [&](size_t bytes) -> float* {
    float* p = (float*)(ws + off);
    off += (bytes + 255) & ~(size_t)255;
    return p;
  };
  float* bufA   = walloc((size_t)NNODES * 128 * 4);
  float* bufB   = walloc((size_t)NNODES * 128 * 4);
  float* dinv   = walloc((size_t)NNODES * 4);
  float* pooled = walloc((size_t)NGRAPH * 64 * 4);
  float* cnt    = walloc((size_t)NGRAPH * 4);
  float* gA     = walloc((size_t)NGRAPH * 128 * 4);
  float* gB     = walloc((size_t)NGRAPH * 64 * 4);
  (void)in_sizes; (void)n_in; (void)out_size; (void)ws_size;

  const int T = 256;
  auto nb = [](long long n) { return dim3((unsigned)((n + 255) / 256)); };

  // --- degrees -> dinv (deg starts at 1.0 for the self loop) ---
  fill_f32<<<nb(NNODES), T, 0, stream>>>(dinv, NNODES, 1.0f);
  edge_deg<<<nb(NEDGES), T, 0, stream>>>(ei, NEDGES, dinv);
  deg_to_dinv<<<nb(NNODES), T, 0, stream>>>(dinv, NNODES);

  // --- one GCN layer: gemm -> self term -> edge scatter -> bias+relu ---
  // M = 50000 is a multiple of 16 -> BCHK=false (no row guards in the hot loop)
  auto gcn = [&](const float* Xin, const float* W, const float* b, int K, int N) {
    int Kp = (K + 3) & ~3;
    dim3 g((NNODES + 127) / 128);
    size_t shm = (size_t)Kp * N * 4;
    if (N == 64)
      gemm_wmma_f32<4, false><<<g, T, shm, stream>>>(Xin, W, nullptr, bufA, NNODES, K, 0);
    else
      gemm_wmma_f32<8, false><<<g, T, shm, stream>>>(Xin, W, nullptr, bufA, NNODES, K, 0);
    self_init<<<nb((long long)NNODES * N), T, 0, stream>>>(bufA, dinv, bufB, NNODES, N);
    edge_scatter<<<nb((long long)NEDGES * (N / 4)), T, 0, stream>>>(ei, dinv, bufA, bufB, NEDGES, N);
    bias_relu<<<nb((long long)NNODES * N), T, 0, stream>>>(bufB, b, NNODES, N);
  };

  gcn(x,    W1, b1,  37,  64);   // bufB <- layer1 [50000 x 64]
  gcn(bufB, W2, b2,  64, 128);   // bufB <- layer2 [50000 x 128]
  gcn(bufB, W3, b3, 128,  64);   // bufB <- layer3 [50000 x 64]

  // --- global mean pool ---
  fill_f32<<<nb(NGRAPH * 64), T, 0, stream>>>(pooled, NGRAPH * 64, 0.0f);
  fill_f32<<<nb(NGRAPH), T, 0, stream>>>(cnt, NGRAPH, 0.0f);
  pool_sum<<<nb((long long)NNODES * 64), T, 0, stream>>>(bufB, batch, pooled, NNODES);
  pool_cnt<<<nb(NNODES), T, 0, stream>>>(batch, cnt, NNODES);
  pool_div<<<nb(NGRAPH * 64), T, 0, stream>>>(pooled, cnt, NGRAPH);

  // --- FC head (bias+relu fused into gemm; M=1000 needs row guards) ---
  {
    dim3 g((NGRAPH + 127) / 128);
    gemm_wmma_f32<8, true><<<g, T, (size_t)64 * 128 * 4, stream>>>(pooled, Wf1, bf1, gA, NGRAPH, 64, 1);
    gemm_wmma_f32<4, true><<<g, T, (size_t)128 * 64 * 4, stream>>>(gA, Wf2, bf2, gB, NGRAPH, 128, 1);
  }

  // --- LayerNorm ---
  layernorm_k<<<nb(NGRAPH), T, 0, stream>>>(gB, gamma, beta, out, NGRAPH);
}